// GroupLorentzConv2d_12824772346234
// MI455X (gfx1250) — compile-verified
//
#include <hip/hip_runtime.h>
#include <math.h>

typedef float v2f __attribute__((ext_vector_type(2)));
typedef float v8f __attribute__((ext_vector_type(8)));

#define KS 3
#define KK 9
#define ICM1 64            // spatial input channels (ic-1)
#define OCM1 64            // spatial output channels (oc-1)
#define OSTABG 4
#define HH 64
#define WWI 64
#define CCH 65
#define KDIM 576           // (ic-1)*KK

// LDS strides
#define XT_WS 66           // per-pixel stride for x tile (64 ch + 2 pad), even -> 8B frag reads
#define XT_HS (34 * XT_WS)
#define BC_S 68            // per-out-channel stride for B chunk; mult of 4 -> 16B async dst align

#define XT_ELEMS (3 * 34 * ICM1)           // 6528
#define XT_ITERS ((XT_ELEMS + 255) / 256)  // 26

__device__ __forceinline__ unsigned lds_lo32(const void* p) {
  // flat->LDS addressing uses addr[31:0] (ISA 10.2): low 32 bits = LDS byte offset
  return (unsigned)(unsigned long long)p;
}

// Async copy of 64 contiguous bytes: INST_OFFSET applies to BOTH the LDS dst
// and the global src (ISA 08 async pseudocode), so one base pair + one LDS
// address VGPR covers all four B128 strips.
__device__ __forceinline__ void async_copy64B(unsigned lds_dst, const float* gsrc) {
  asm volatile(
      "global_load_async_to_lds_b128 %0, %1, off\n\t"
      "global_load_async_to_lds_b128 %0, %1, off offset:16\n\t"
      "global_load_async_to_lds_b128 %0, %1, off offset:32\n\t"
      "global_load_async_to_lds_b128 %0, %1, off offset:48"
      :: "v"(lds_dst), "v"(gsrc) : "memory");
}

// ---------------------------------------------------------------------------
// Prep: Wsp[g][o][m] = rot90^g(weight_space[o][m/9])[m%9]
// ---------------------------------------------------------------------------
__global__ __launch_bounds__(256) void glconv_prep(const float* __restrict__ wsp,
                                                   float* __restrict__ Wb) {
  int idx = blockIdx.x * 256 + threadIdx.x;
  if (idx >= OSTABG * OCM1 * KDIM) return;
  int m = idx % KDIM;
  int o = (idx / KDIM) % OCM1;
  int g = idx / (KDIM * OCM1);
  int cw = m / KK, kw = m % KK;
  int i = kw / KS, j = kw % KS;
  int si, sj;
  if (g == 0)      { si = i;        sj = j;        }
  else if (g == 1) { si = j;        sj = KS-1 - i; }
  else if (g == 2) { si = KS-1 - i; sj = KS-1 - j; }
  else             { si = KS-1 - j; sj = i;        }
  Wb[idx] = wsp[(o * ICM1 + cw) * KK + si * KS + sj];
}

// ---------------------------------------------------------------------------
// Fused im2col + fp32 WMMA GEMM + Lorentz epilogue.
// Block: 32 pixels (same b,h) x 64 out channels, one rotation group g.
// B chunks double-buffered in LDS via GLOBAL_LOAD_ASYNC_TO_LDS_B128.
// ---------------------------------------------------------------------------
__global__ __launch_bounds__(256) void glconv_gemm(const float* __restrict__ x,
                                                   const float* __restrict__ Wb,
                                                   const float* __restrict__ wti,
                                                   float* __restrict__ out) {
  __shared__ float xt[3 * XT_HS];        // 26.9 KB spatial halo tile
  __shared__ float tt[3 * 34];           // time channel halo
  __shared__ float Bc[2][OCM1 * BC_S];   // 2 x 17.4 KB double-buffered B chunk
  __shared__ float tl[32];
  __shared__ float part[4][32];

  const int tid   = threadIdx.x;
  const int pb    = blockIdx.x;
  const int g     = blockIdx.y;
  const int b     = pb >> 7;
  const int lbase = (pb & 127) << 5;
  const int h     = lbase >> 6;
  const int w0    = lbase & 63;

  // --- async B prefetch: thread -> (row o = tid/4, 16-float strip) ---
  const int bo = tid >> 2;
  const int bc = (tid & 3) << 4;
  const float* bsrc0  = Wb + (size_t)(g * OCM1 + bo) * KDIM + bc;
  const unsigned bdst0 = lds_lo32(&Bc[0][bo * BC_S + bc]);
  const unsigned bdst1 = lds_lo32(&Bc[1][bo * BC_S + bc]);

  // issue chunk 0 into buffer 0 (overlaps with x staging below)
  async_copy64B(bdst0, bsrc0);

  // --- stage x halo tile: batched loads into registers, then LDS stores ---
  float xv[XT_ITERS];
  #pragma unroll
  for (int it = 0; it < XT_ITERS; ++it) {
    int idx = tid + 256 * it;
    int c = idx & 63, p = idx >> 6;
    int wwi = p % 34, hhi = p / 34;
    int gh = h + hhi - 1, gw = w0 + wwi - 1;
    bool ib = (idx < XT_ELEMS) && gh >= 0 && gh < HH && gw >= 0 && gw < WWI;
    size_t off = ib ? ((size_t)(b * HH + gh) * WWI + gw) * CCH + 1 + c : 0;
    xv[it] = ib ? x[off] : 0.0f;
  }
  #pragma unroll
  for (int it = 0; it < XT_ITERS; ++it) {
    int idx = tid + 256 * it;
    if (idx < XT_ELEMS) {
      int c = idx & 63, p = idx >> 6;
      int wwi = p % 34, hhi = p / 34;
      xt[hhi * XT_HS + wwi * XT_WS + c] = xv[it];
    }
  }
  // --- time channel (102 elements, one shot) ---
  if (tid < 3 * 34) {
    int wwi = tid % 34, hhi = tid / 34;
    int gh = h + hhi - 1, gw = w0 + wwi - 1;
    float v = 0.0f;
    if (gh >= 0 && gh < HH && gw >= 0 && gw < WWI)
      v = x[((size_t)(b * HH + gh) * WWI + gw) * CCH];
    tt[tid] = v;
  }
  __syncthreads();

  if (tid < 32) {
    float s = 0.0f;
    #pragma unroll
    for (int kk = 0; kk < KK; kk++) {
      int i = kk / 3, j = kk % 3;
      float pt = fmaxf(tt[i * 34 + tid + j], 1.0f);  // OOB stored 0 -> 1
      s += pt * pt;
    }
    tl[tid] = sqrtf(s - (float)(KK - 1));
  }

  const int wid  = tid >> 5;
  const int wm   = wid >> 2;       // 0..1 : pixel-row subtile
  const int wn   = wid & 3;        // 0..3 : out-channel subtile
  const int lane = tid & 31;
  const int lh   = lane & 15;
  const int hi   = lane >> 4;      // K half (0: K0/K1, 1: K2/K3)

  v8f acc = {};

  for (int kc = 0; kc < KK; kc++) {
    if (kc + 1 < KK) {
      // prefetch next chunk into the other buffer (last read 2 iterations ago)
      async_copy64B((kc + 1) & 1 ? bdst1 : bdst0, bsrc0 + (kc + 1) * 64);
      asm volatile("s_wait_asynccnt 4" ::: "memory");  // chunk kc landed
    } else {
      asm volatile("s_wait_asynccnt 0" ::: "memory");
    }
    __syncthreads();   // chunk kc visible to all waves

    const int ki = kc / 3, kj = kc % 3;
    // A[M][K]: lane lh -> row M=wm*16+lh, VGPRs hold K = hi*2, hi*2+1
    const float* abase = &xt[ki * XT_HS + (wm * 16 + lh + kj) * XT_WS + hi * 2];
    // B[K][N]: lane lh -> col N=wn*16+lh, VGPRs hold K = hi*2, hi*2+1
    const float* bbase = &Bc[kc & 1][(wn * 16 + lh) * BC_S + hi * 2];

    #pragma unroll
    for (int k4 = 0; k4 < 16; k4++) {
      v2f a  = *(const v2f*)(abase + k4 * 4);
      v2f bf = *(const v2f*)(bbase + k4 * 4);
      acc = __builtin_amdgcn_wmma_f32_16x16x4_f32(false, a, false, bf,
                                                  (short)0, acc, false, false);
    }
    __syncthreads();   // buffer kc&1 free for reuse at chunk kc+2
  }

  // --- epilogue: time rank-1 term, write channels 1..64, square-reduce ---
  const int   ocol     = wn * 16 + lh;
  const float wv       = wti[1 + ocol];
  const size_t pixBase = (size_t)(b * OSTABG + g) * (HH * WWI) + lbase;

  #pragma unroll
  for (int v = 0; v < 8; v++) {
    int rl = wm * 16 + hi * 8 + v;                 // C layout: M = v + hi*8
    float val = acc[v] + tl[rl] * wv;
    out[(pixBase + rl) * CCH + 1 + ocol] = val;
    float s = val * val;
    s += __shfl_xor(s, 1);                          // reduce within 16-lane half
    s += __shfl_xor(s, 2);
    s += __shfl_xor(s, 4);
    s += __shfl_xor(s, 8);
    if (lh == 0) part[wn][rl] = s;
  }
  __syncthreads();

  // --- channel 0: ot = sqrt(1 + sum over 64 channels of val^2) ---
  if (tid < 32) {
    float tot = part[0][tid] + part[1][tid] + part[2][tid] + part[3][tid];
    out[(pixBase + tid) * CCH] = sqrtf(1.0f + tot);
  }
}

// ---------------------------------------------------------------------------
extern "C" void kernel_launch(void* const* d_in, const int* in_sizes, int n_in,
                              void* d_out, int out_size, void* d_ws, size_t ws_size,
                              hipStream_t stream) {
  (void)in_sizes; (void)n_in; (void)out_size; (void)ws_size;
  const float* x   = (const float*)d_in[0];   // (16,64,64,65)
  const float* wsp = (const float*)d_in[1];   // (64,64,1,3,3)
  // d_in[2] (w_time_output) is dead in the reference output
  const float* wti = (const float*)d_in[3];   // (65,1)
  float* Wb  = (float*)d_ws;                  // 4*64*576 floats = 590 KB
  float* out = (float*)d_out;                 // (16,4,64,64,65)

  glconv_prep<<<(OSTABG * OCM1 * KDIM + 255) / 256, 256, 0, stream>>>(wsp, Wb);
  dim3 grid(2048, OSTABG);
  glconv_gemm<<<grid, 256, 0, stream>>>(x, Wb, wti, out);
}